// DASBeamformLayer_85109071937545
// MI455X (gfx1250) — compile-verified
//
#include <hip/hip_runtime.h>
#include <stdint.h>

// DAS beamforming, CDNA5 (gfx1250).
// Strategy: per block, loop over 128 channels; stage each 32KB rf[b,c] slice
// into LDS with the Tensor Data Mover (one tensor_load_to_lds per channel,
// double buffered, s_wait_tensorcnt + barrier), then every thread does the
// delay + 2-tap lerp gather from LDS (ds_load_b128) for its 4 pixels.

namespace {
constexpr int kB  = 2;
constexpr int kNc = 128;
constexpr int kNs = 2048;
constexpr int kNz = 256;
constexpr int kNx = 256;
constexpr int kK  = 4;
constexpr int kM  = kNz * kNx;             // 65536 pixels per batch
constexpr int kThreads = 256;              // 8 wave32
constexpr int kPPT = 4;                    // pixels per thread
constexpr int kTile = kThreads * kPPT;     // 1024 pixels per block
constexpr int kTilesPerBatch = kM / kTile; // 64
constexpr int kChanFloats = kNs * kK;      // 8192 floats = 32 KB
constexpr uint32_t kBufBytes = (uint32_t)kChanFloats * 4u;
}

typedef unsigned int u32x4 __attribute__((ext_vector_type(4)));
typedef int i32x8 __attribute__((ext_vector_type(8)));
typedef int i32x4 __attribute__((ext_vector_type(4)));

// Issue one TDM transfer: 8192 x f32 (one rf[b,c] slice) -> LDS.
// D# per CDNA5 ISA ch.8: group0 = count/lds_addr/global_addr/type,
// group1 = data_size + dims/strides, groups 2-3 unused (2-D tensor).
__device__ __forceinline__ void tdm_stage_channel(const float* src,
                                                  uint32_t lds_byte_addr) {
  const uint64_t ga = (uint64_t)(uintptr_t)src;
  constexpr uint32_t td0 = (uint32_t)kChanFloats;  // 8192 elements of 4B
  u32x4 g0;
  g0.x = 1u;                                        // count=1, user descriptor
  g0.y = lds_byte_addr;                             // lds_addr[31:0]
  g0.z = (uint32_t)ga;                              // global_addr[31:0]
  g0.w = ((uint32_t)(ga >> 32) & 0x01FFFFFFu)       // global_addr[56:32]
         | (2u << 30);                              // type = 2 ("image")
  i32x8 g1 = {
      (int)(2u << 16),                              // data_size = 2 -> 4 bytes
      (int)((td0 & 0xFFFFu) << 16),                 // tensor_dim0[15:0]
      (int)(((td0 >> 16) & 0xFFFFu) | (1u << 16)),  // dim0[31:16] | tensor_dim1=1
      (int)((td0 & 0xFFFFu) << 16),                 // dim1[31:16]=0 | tile_dim0=8192
      1,                                            // tile_dim1=1, tile_dim2=0
      (int)td0,                                     // tensor_dim0_stride[31:0]
      0,                                            // stride0 hi / stride1 lo
      0};                                           // stride1 hi
  i32x4 gz4 = {0, 0, 0, 0};
  i32x8 gz8 = {0, 0, 0, 0, 0, 0, 0, 0};
  __builtin_amdgcn_tensor_load_to_lds(g0, g1, gz4, gz4, gz8, /*cpol=*/0);
}

__global__ void __launch_bounds__(kThreads)
das_beamform_kernel(const float* __restrict__ rf, const float* __restrict__ g,
                    const float* __restrict__ pr, const float* __restrict__ p,
                    float* __restrict__ out) {
  extern __shared__ float smem[];  // 2 buffers x 8192 floats = 64 KB

  const int tid  = (int)threadIdx.x;
  const int b    = (int)blockIdx.x / kTilesPerBatch;
  const int tile = (int)blockIdx.x % kTilesPerBatch;

  // Per-batch acquisition params (uniform -> SGPR).
  const float c0 = p[b * 4 + 0];
  const float fs = p[b * 4 + 1];
  const float t0 = p[b * 4 + 2];
  const float scale = fs / c0;         // samples per meter
  const float base  = fs * t0 / c0;    // constant sample offset

  // This thread's pixels: coordinates + transmit path (= depth z).
  int pix[kPPT];
  float gx[kPPT], gy[kPPT], gzc[kPPT];
  float4 acc[kPPT];
#pragma unroll
  for (int q = 0; q < kPPT; ++q) {
    pix[q] = tile * kTile + q * kThreads + tid;
    const float* gp = g + ((size_t)b * kM + (size_t)pix[q]) * 3;
    gx[q] = gp[0];
    gy[q] = gp[1];
    gzc[q] = gp[2];
    acc[q] = make_float4(0.f, 0.f, 0.f, 0.f);
  }

  // Low 32 bits of the generic LDS pointer == byte offset in LDS space.
  const uint32_t lds_base = (uint32_t)(uintptr_t)smem;
  const float* rf_b = rf + (size_t)b * kNc * kChanFloats;

  // Prologue: stage channel 0 into buffer 0 (wave 0 only; TDM ignores EXEC,
  // so this must be a real branch).
  if (tid < 32) {
    tdm_stage_channel(rf_b, lds_base);
  }

  for (int c = 0; c < kNc; ++c) {
    if (tid < 32) {
      if (c + 1 < kNc) {
        // Double-buffer: prefetch next channel, then wait until only that
        // one transfer is still in flight (TDM retires in order).
        tdm_stage_channel(rf_b + (size_t)(c + 1) * kChanFloats,
                          lds_base + (uint32_t)((c + 1) & 1) * kBufBytes);
        __builtin_amdgcn_s_wait_tensorcnt(1);
      } else {
        __builtin_amdgcn_s_wait_tensorcnt(0);
      }
    }
    __syncthreads();  // buffer (c&1) now holds rf[b,c]

    // Receiver element position (uniform scalar loads).
    const float prx = pr[((size_t)b * kNc + c) * 3 + 0];
    const float pry = pr[((size_t)b * kNc + c) * 3 + 1];
    const float prz = pr[((size_t)b * kNc + c) * 3 + 2];

    const float4* s4 =
        reinterpret_cast<const float4*>(smem + (c & 1) * kChanFloats);

#pragma unroll
    for (int q = 0; q < kPPT; ++q) {
      const float dx = gx[q] - prx;
      const float dy = gy[q] - pry;
      const float dz = gzc[q] - prz;
      const float d_rx = sqrtf(dx * dx + dy * dy + dz * dz);
      float s = scale * (d_rx + gzc[q]) + base;           // fractional sample
      s = fminf(fmaxf(s, 0.0f), (float)(kNs - 1));        // clamp extension
      int i0 = (int)s;                                    // floor (s >= 0)
      if (i0 > kNs - 2) i0 = kNs - 2;
      const float w  = s - (float)i0;
      const float w1 = 1.0f - w;
      const float4 y0 = s4[i0];        // ds_load_b128
      const float4 y1 = s4[i0 + 1];    // ds_load_b128
      acc[q].x = fmaf(w1, y0.x, fmaf(w, y1.x, acc[q].x));
      acc[q].y = fmaf(w1, y0.y, fmaf(w, y1.y, acc[q].y));
      acc[q].z = fmaf(w1, y0.z, fmaf(w, y1.z, acc[q].z));
      acc[q].w = fmaf(w1, y0.w, fmaf(w, y1.w, acc[q].w));
    }
    __syncthreads();  // all reads of buffer (c&1) done before it is restaged
  }

#pragma unroll
  for (int q = 0; q < kPPT; ++q) {
    float4* op =
        reinterpret_cast<float4*>(out + ((size_t)b * kM + (size_t)pix[q]) * kK);
    *op = acc[q];
  }
}

extern "C" void kernel_launch(void* const* d_in, const int* in_sizes, int n_in,
                              void* d_out, int out_size, void* d_ws,
                              size_t ws_size, hipStream_t stream) {
  (void)in_sizes; (void)n_in; (void)out_size; (void)d_ws; (void)ws_size;
  const float* rf = (const float*)d_in[0];
  const float* g  = (const float*)d_in[1];
  const float* pr = (const float*)d_in[2];
  const float* p  = (const float*)d_in[3];
  float* out = (float*)d_out;

  dim3 grid(kB * kTilesPerBatch);   // 128 blocks
  dim3 block(kThreads);             // 256 threads = 8 wave32
  size_t shmem = 2u * (size_t)kChanFloats * sizeof(float);  // 64 KB
  das_beamform_kernel<<<grid, block, shmem, stream>>>(rf, g, pr, p, out);
}